// MappingToContinuous_86981677678839
// MI455X (gfx1250) — compile-verified
//
#include <hip/hip_runtime.h>
#include <math.h>
#include <stdint.h>

#define KDIM 128
#define WAVES_PER_BLOCK 8
#define ROWS_PER_WAVE 4
#define ROWS_PER_BLOCK (WAVES_PER_BLOCK * ROWS_PER_WAVE) // 32

typedef float v4f __attribute__((ext_vector_type(4)));

// ---------------- label relabeling (unique -> inverse) ----------------------
__global__ void tn_zero_present(int* __restrict__ present) {
  present[threadIdx.x] = 0;
}

__global__ void tn_mark_present(const int* __restrict__ C,
                                int* __restrict__ present, int n) {
  int i = blockIdx.x * blockDim.x + threadIdx.x;
  if (i < n) present[C[i]] = 1; // benign race: all writers store 1
}

__global__ void tn_build_rank(const int* __restrict__ present,
                              int* __restrict__ rank) {
  if (threadIdx.x == 0) {
    int acc = 0;
    for (int v = 0; v < KDIM; ++v) {
      rank[v] = acc;       // exclusive prefix = index among sorted uniques
      acc += present[v];
    }
  }
}

// ---------------- fast float32 ndtri via Giles erfinv -----------------------
// ndtri(p) = sqrt(2) * erfinv(2p-1);  w = -ln(4 p (1-p)) == -ln(1-x^2)
__device__ __forceinline__ float ndtri_fast(float p) {
  float x = fmaf(2.0f, p, -1.0f);
  float t = 4.0f * p * (1.0f - p);     // exact form in p: no cancellation at p->0
  float w = -__logf(t);
  // central branch (w < 5)
  float wc = w - 2.5f;
  float pc = 2.81022636e-08f;
  pc = fmaf(pc, wc, 3.43273939e-07f);
  pc = fmaf(pc, wc, -3.5233877e-06f);
  pc = fmaf(pc, wc, -4.39150654e-06f);
  pc = fmaf(pc, wc, 0.00021858087f);
  pc = fmaf(pc, wc, -0.00125372503f);
  pc = fmaf(pc, wc, -0.00417768164f);
  pc = fmaf(pc, wc, 0.246640727f);
  pc = fmaf(pc, wc, 1.50140941f);
  // tail branch (w >= 5), covers p down to ~1e-30
  float wt = sqrtf(fmaxf(w, 0.0f)) - 3.0f;
  float pt = -0.000200214257f;
  pt = fmaf(pt, wt, 0.000100950558f);
  pt = fmaf(pt, wt, 0.00134934322f);
  pt = fmaf(pt, wt, -0.00367342844f);
  pt = fmaf(pt, wt, 0.00573950773f);
  pt = fmaf(pt, wt, -0.0076224613f);
  pt = fmaf(pt, wt, 0.00943887047f);
  pt = fmaf(pt, wt, 1.00167406f);
  pt = fmaf(pt, wt, 2.83297682f);
  float r = (w < 5.0f) ? pc : pt;      // branchless select, no divergence
  return 1.41421356237f * r * x;
}

// ---------------- main streaming kernel -------------------------------------
// One wave per row: 32 lanes x 4 floats = 128 = K. Each wave owns 4 rows with
// a 4-deep LDS pipeline filled by gfx1250 async global->LDS DMA. All 4 DMAs
// are issued up-front (max MLP), drained with s_wait_asynccnt 3/2/1/0 using
// the ISA's in-order async completion. Producer wave == consumer wave, so no
// barriers are needed.
__global__ __launch_bounds__(256) void tn_truncnorm_kernel(
    const int* __restrict__ C, const float* __restrict__ eps,
    const float* __restrict__ U, const float* __restrict__ mu_p,
    const float* __restrict__ sigma_p, const int* __restrict__ rank,
    float* __restrict__ out) {
  __shared__ __align__(16)
      float sBuf[WAVES_PER_BLOCK * ROWS_PER_WAVE * KDIM]; // 16 KB

  const int lane = threadIdx.x & 31;
  const int wave = threadIdx.x >> 5;
  const float mu = mu_p[0];
  const float sigma = sigma_p[0];

  const long long rowBase =
      (long long)blockIdx.x * ROWS_PER_BLOCK + (long long)wave * ROWS_PER_WAVE;

  // issue all 4 row DMAs up-front: U is a read-once 128 MiB stream -> NT hint
#pragma unroll
  for (int j = 0; j < ROWS_PER_WAVE; ++j) {
    unsigned ldsOff = (unsigned)(uintptr_t)(void*)
        &sBuf[(wave * ROWS_PER_WAVE + j) * KDIM + lane * 4];
    const float* g = U + (size_t)(rowBase + j) * KDIM + lane * 4;
    asm volatile("global_load_async_to_lds_b128 %0, %1, off th:TH_LOAD_NT"
                 :: "v"(ldsOff), "v"(g) : "memory");
  }

#pragma unroll
  for (int j = 0; j < ROWS_PER_WAVE; ++j) {
    const long long row = rowBase + j;

    // async loads complete in order: <= (3-j) outstanding => row j resident
    if (j == 0)      asm volatile("s_wait_asynccnt 0x3" ::: "memory");
    else if (j == 1) asm volatile("s_wait_asynccnt 0x2" ::: "memory");
    else if (j == 2) asm volatile("s_wait_asynccnt 0x1" ::: "memory");
    else             asm volatile("s_wait_asynccnt 0x0" ::: "memory");

    // per-row scalars (wave-uniform; broadcast load)
    const float u   = fmaf(sigma, eps[row], mu);
    const float phi = 0.5f * erfcf(-u * 0.70710678118654752f); // ndtr(u)
    const int  csel = rank[C[row]];

    const v4f Uv =
        *(const v4f*)&sBuf[(wave * ROWS_PER_WAVE + j) * KDIM + lane * 4];
    float r0[4] = {Uv.x, Uv.y, Uv.z, Uv.w};
    float o[4];
#pragma unroll
    for (int e = 0; e < 4; ++e) {
      float uc = fminf(fmaxf(r0[e], 1e-7f), 1.0f - 1e-7f);
      float p  = uc * phi;
      float z  = ndtri_fast(p);
      int   k  = lane * 4 + e;
      o[e] = (k == csel) ? u : z;
    }
    v4f ov = {o[0], o[1], o[2], o[3]};
    // write-once 128 MiB stream: non-temporal b128 store
    __builtin_nontemporal_store(ov, (v4f*)(out + (size_t)row * KDIM) + lane);
  }
}

// ---------------- launch ----------------------------------------------------
extern "C" void kernel_launch(void* const* d_in, const int* in_sizes, int n_in,
                              void* d_out, int out_size, void* d_ws,
                              size_t ws_size, hipStream_t stream) {
  const int*   C     = (const int*)d_in[0];
  const float* eps   = (const float*)d_in[1];
  const float* U     = (const float*)d_in[2];
  const float* mu    = (const float*)d_in[3];
  const float* sigma = (const float*)d_in[4];
  float*       out   = (float*)d_out;

  const int nRows = in_sizes[0]; // B*N = 262144 ; K fixed at 128

  int* present = (int*)d_ws;         // [128]
  int* rank    = present + KDIM;     // [128]

  tn_zero_present<<<1, KDIM, 0, stream>>>(present);
  tn_mark_present<<<(nRows + 255) / 256, 256, 0, stream>>>(C, present, nRows);
  tn_build_rank<<<1, 32, 0, stream>>>(present, rank);

  tn_truncnorm_kernel<<<nRows / ROWS_PER_BLOCK, 256, 0, stream>>>(
      C, eps, U, mu, sigma, rank, out);
}